// LRULayer_24120536334388
// MI455X (gfx1250) — compile-verified
//
#include <hip/hip_runtime.h>
#include <math.h>

typedef __attribute__((ext_vector_type(2))) float v2f;
typedef __attribute__((ext_vector_type(4))) float v4f;
typedef __attribute__((ext_vector_type(8))) float v8f;

#define TILE_M 128
#define TILE_N 128
#define TILE_K 32
#define A_LDS_STRIDE (TILE_K + 4)   // 36 floats: 16B-aligned rows + bank spread
#define B_LDS_STRIDE TILE_N         // 128 floats

#define NCHUNK 32                   // time chunks for the parallel scan
// CHUNK_LEN = T / NCHUNK = 64 (power of two -> Lam^L by repeated squaring)

// ---------------------------------------------------------------------------
// Generic fp32 GEMM: out[M,Nn] = A[M,K] @ Bm[K,Nn] (+ bias[n]) (+ Dv[n]*U[m,n])
// Workgroup: 256 threads (8 waves). Tile 128x128, K-step 32.
// Each wave: 64x32 patch = 4x2 tiles of 16x16 via V_WMMA_F32_16X16X4_F32.
// All dims must be multiples of the tile sizes (true for this problem).
// ---------------------------------------------------------------------------
__global__ __launch_bounds__(256)
void gemm_f32_wmma(const float* __restrict__ A, const float* __restrict__ Bm,
                   float* __restrict__ out,
                   const float* __restrict__ bias,   // nullable, len Nn
                   const float* __restrict__ Dv,     // nullable, len Nn
                   const float* __restrict__ U,      // nullable, [M,Nn]
                   int M, int K, int Nn) {
  __shared__ float As[TILE_M * A_LDS_STRIDE];
  __shared__ float Bs[TILE_K * B_LDS_STRIDE];

  const int tid  = threadIdx.x;
  const int lane = tid & 31;
  const int wid  = tid >> 5;     // 0..7
  const int wm   = wid >> 2;     // 0..1 : which 64-row half
  const int wn   = wid & 3;      // 0..3 : which 32-col strip
  const int half = lane >> 4;    // 0/1 : K-half within WMMA operand
  const int l16  = lane & 15;

  const int block_m = blockIdx.y * TILE_M;
  const int block_n = blockIdx.x * TILE_N;

  v8f acc[4][2];
  const v8f vzero = {};
#pragma unroll
  for (int i = 0; i < 4; ++i)
#pragma unroll
    for (int j = 0; j < 2; ++j) acc[i][j] = vzero;

  for (int k0 = 0; k0 < K; k0 += TILE_K) {
    // ---- stage A tile [128 x 32] : 1024 float4, 4 per thread ----
    {
      const float* gA = A + (size_t)block_m * K + k0;
#pragma unroll
      for (int i = 0; i < 4; ++i) {
        const int idx = tid + i * 256;        // 0..1023
        const int row = idx >> 3;             // 0..127
        const int c4  = (idx & 7) << 2;       // 0..28 step 4
        v4f v = *(const v4f*)(gA + (size_t)row * K + c4);
        *(v4f*)(&As[row * A_LDS_STRIDE + c4]) = v;
      }
      // ---- stage B tile [32 x 128] : 1024 float4, 4 per thread ----
      const float* gB = Bm + (size_t)k0 * Nn + block_n;
#pragma unroll
      for (int i = 0; i < 4; ++i) {
        const int idx = tid + i * 256;
        const int row = idx >> 5;             // 0..31
        const int c4  = (idx & 31) << 2;      // 0..124 step 4
        v4f v = *(const v4f*)(gB + (size_t)row * Nn + c4);
        *(v4f*)(&Bs[row * B_LDS_STRIDE + c4]) = v;
      }
      // ---- prefetch next K tile into cache while WMMAs run ----
      if (k0 + TILE_K < K) {
        const float* pA = A + (size_t)block_m * K + (k0 + TILE_K);
        const float* pB = Bm + (size_t)(k0 + TILE_K) * Nn + block_n;
        if (tid < 128) {
          // A: 128 rows x 128B each
          __builtin_prefetch(pA + (size_t)tid * K, 0, 3);
        } else {
          // B: 32 rows x 512B each -> 4 segments of 128B
          const int idx = tid - 128;
          const int row = idx >> 2;
          const int seg = idx & 3;
          __builtin_prefetch(pB + (size_t)row * Nn + seg * 32, 0, 3);
        }
      }
    }
    __syncthreads();

#pragma unroll
    for (int kk = 0; kk < TILE_K; kk += 4) {
      const int kidx = kk + half * 2;
      v2f a[4], b[2];
#pragma unroll
      for (int i = 0; i < 4; ++i) {
        const int m = wm * 64 + i * 16 + l16;
        a[i] = *(const v2f*)(&As[m * A_LDS_STRIDE + kidx]);
      }
#pragma unroll
      for (int j = 0; j < 2; ++j) {
        const int n = wn * 32 + j * 16 + l16;
        v2f t;
        t.x = Bs[kidx * B_LDS_STRIDE + n];
        t.y = Bs[(kidx + 1) * B_LDS_STRIDE + n];
        b[j] = t;
      }
#pragma unroll
      for (int i = 0; i < 4; ++i)
#pragma unroll
        for (int j = 0; j < 2; ++j)
          acc[i][j] = __builtin_amdgcn_wmma_f32_16x16x4_f32(
              false, a[i], false, b[j], (short)0, acc[i][j], false, false);
    }
    __syncthreads();
  }

  // ---- epilogue: D layout -> lane l16 = col, vgpr v + 8*half = row ----
#pragma unroll
  for (int i = 0; i < 4; ++i) {
#pragma unroll
    for (int j = 0; j < 2; ++j) {
      const int gn = block_n + wn * 32 + j * 16 + l16;
      const float bval = bias ? bias[gn] : 0.0f;
      const float dval = Dv ? Dv[gn] : 0.0f;
#pragma unroll
      for (int v = 0; v < 8; ++v) {
        const int gm = block_m + wm * 64 + i * 16 + half * 8 + v;
        float r = acc[i][j][v] + bval;
        if (Dv) r += dval * U[(size_t)gm * Nn + gn];
        out[(size_t)gm * Nn + gn] = r;
      }
    }
  }
}

// ---------------------------------------------------------------------------
// Lam = exp(-exp(nu_log) + i*exp(theta_log))
// ---------------------------------------------------------------------------
__global__ void prep_lam(const float* __restrict__ nu_log,
                         const float* __restrict__ theta_log,
                         float* __restrict__ lam_r, float* __restrict__ lam_i,
                         int Nc) {
  int n = blockIdx.x * blockDim.x + threadIdx.x;
  if (n < Nc) {
    float mag = expf(-expf(nu_log[n]));
    float th  = expf(theta_log[n]);
    lam_r[n] = mag * cosf(th);
    lam_i[n] = mag * sinf(th);
  }
}

// Bcat[k, j] (k<N, j<2N): j<N -> B_real[k,j]*e^gamma[k] ; else B_imag[k,j-N]*e^gamma[k]
__global__ void prep_Bcat(const float* __restrict__ B_real,
                          const float* __restrict__ B_imag,
                          const float* __restrict__ gamma_log,
                          float* __restrict__ Bcat, int Nc) {
  int idx = blockIdx.x * blockDim.x + threadIdx.x;
  int total = Nc * 2 * Nc;
  if (idx < total) {
    int k = idx / (2 * Nc);
    int j = idx - k * 2 * Nc;
    float g = expf(gamma_log[k]);
    float v = (j < Nc) ? B_real[(size_t)k * Nc + j] : B_imag[(size_t)k * Nc + (j - Nc)];
    Bcat[idx] = v * g;
  }
}

// Ccat[k, j] (k<2N, j<N): k<N -> C_real[k,j] ; else -C_imag[k-N,j]
__global__ void prep_Ccat(const float* __restrict__ C_real,
                          const float* __restrict__ C_imag,
                          float* __restrict__ Ccat, int Nc) {
  int idx = blockIdx.x * blockDim.x + threadIdx.x;
  int total = 2 * Nc * Nc;
  if (idx < total) {
    int k = idx / Nc;
    int j = idx - k * Nc;
    Ccat[idx] = (k < Nc) ? C_real[(size_t)k * Nc + j]
                         : -C_imag[(size_t)(k - Nc) * Nc + j];
  }
}

// ---------------------------------------------------------------------------
// Parallel chunked scan over time. Layout [T*B, 2N]: row=t*B+b, col n real,
// col N+n imag. Since Lam is constant per channel, the scan decomposes:
//   P1: per-chunk end state with zero init (read-only)
//   P2: serial combine over NCHUNK chunk ends using Lam^L
//   P3: per-chunk re-walk from correct start state, writing h in place
// ---------------------------------------------------------------------------
__global__ void scan_partial(const float* __restrict__ Bu,
                             const float* __restrict__ lam_r,
                             const float* __restrict__ lam_i,
                             float* __restrict__ E,   // [NCHUNK*B, 2N]
                             int T, int Bb, int Nc) {
  const int idx = blockIdx.x * blockDim.x + threadIdx.x;  // (c*B + b)*N + n
  if (idx >= NCHUNK * Bb * Nc) return;
  const int n  = idx % Nc;
  const int cb = idx / Nc;          // c*B + b
  const int b  = cb % Bb;
  const int c  = cb / Bb;
  const int L  = T / NCHUNK;
  const float lr = lam_r[n], li = lam_i[n];
  float hr = 0.0f, hi = 0.0f;
  const size_t rowStride = (size_t)Bb * 2 * Nc;
  const float* p = Bu + ((size_t)(c * L) * Bb + b) * 2 * Nc + n;
  for (int t = 0; t < L; ++t) {
    float br = p[0];
    float bi = p[Nc];
    float nhr = fmaf(lr, hr, fmaf(-li, hi, br));
    float nhi = fmaf(lr, hi, fmaf(li, hr, bi));
    hr = nhr; hi = nhi;
    p += rowStride;
  }
  E[(size_t)cb * 2 * Nc + n]      = hr;
  E[(size_t)cb * 2 * Nc + Nc + n] = hi;
}

__global__ void scan_starts(const float* __restrict__ E,  // [NCHUNK*B, 2N]
                            const float* __restrict__ lam_r,
                            const float* __restrict__ lam_i,
                            float* __restrict__ S,        // [NCHUNK*B, 2N]
                            int T, int Bb, int Nc) {
  const int idx = blockIdx.x * blockDim.x + threadIdx.x;  // b*N + n
  if (idx >= Bb * Nc) return;
  const int n = idx % Nc;
  const int b = idx / Nc;
  const int L = T / NCHUNK;
  // lamL = Lam^L via repeated squaring (L is a power of two)
  float ar = lam_r[n], ai = lam_i[n];
  for (int l = L; l > 1; l >>= 1) {
    float nr = ar * ar - ai * ai;
    float ni = 2.0f * ar * ai;
    ar = nr; ai = ni;
  }
  float sr = 0.0f, si = 0.0f;
  for (int c = 0; c < NCHUNK; ++c) {
    const size_t off = ((size_t)c * Bb + b) * 2 * Nc + n;
    S[off]      = sr;
    S[off + Nc] = si;
    const float er = E[off];
    const float ei = E[off + Nc];
    float nsr = fmaf(ar, sr, fmaf(-ai, si, er));
    float nsi = fmaf(ar, si, fmaf(ai, sr, ei));
    sr = nsr; si = nsi;
  }
}

__global__ void scan_final(float* __restrict__ BuH,       // in/out [T*B, 2N]
                           const float* __restrict__ lam_r,
                           const float* __restrict__ lam_i,
                           const float* __restrict__ S,   // [NCHUNK*B, 2N]
                           int T, int Bb, int Nc) {
  const int idx = blockIdx.x * blockDim.x + threadIdx.x;  // (c*B + b)*N + n
  if (idx >= NCHUNK * Bb * Nc) return;
  const int n  = idx % Nc;
  const int cb = idx / Nc;
  const int b  = cb % Bb;
  const int c  = cb / Bb;
  const int L  = T / NCHUNK;
  const float lr = lam_r[n], li = lam_i[n];
  float hr = S[(size_t)cb * 2 * Nc + n];
  float hi = S[(size_t)cb * 2 * Nc + Nc + n];
  const size_t rowStride = (size_t)Bb * 2 * Nc;
  float* p = BuH + ((size_t)(c * L) * Bb + b) * 2 * Nc + n;
  for (int t = 0; t < L; ++t) {
    float br = p[0];
    float bi = p[Nc];
    float nhr = fmaf(lr, hr, fmaf(-li, hi, br));
    float nhi = fmaf(lr, hi, fmaf(li, hr, bi));
    hr = nhr; hi = nhi;
    p[0]  = hr;
    p[Nc] = hi;
    p += rowStride;
  }
}

extern "C" void kernel_launch(void* const* d_in, const int* in_sizes, int n_in,
                              void* d_out, int out_size, void* d_ws, size_t ws_size,
                              hipStream_t stream) {
  const float* x         = (const float*)d_in[0];
  const float* W_in      = (const float*)d_in[1];
  const float* b_in      = (const float*)d_in[2];
  const float* W_out     = (const float*)d_in[3];
  const float* b_out     = (const float*)d_in[4];
  const float* nu_log    = (const float*)d_in[5];
  const float* theta_log = (const float*)d_in[6];
  const float* gamma_log = (const float*)d_in[7];
  const float* B_real    = (const float*)d_in[8];
  const float* B_imag    = (const float*)d_in[9];
  const float* C_real    = (const float*)d_in[10];
  const float* C_imag    = (const float*)d_in[11];
  const float* Dvec      = (const float*)d_in[12];

  const int T = 2048, Bb = 8, E = 512, N = 1024;
  const int M = T * Bb;  // 16384

  // Workspace layout (floats)
  float* ws    = (float*)d_ws;
  float* u     = ws;                            // M*N       (64 MiB)
  float* BuH   = u    + (size_t)M * N;          // M*2N      (128 MiB) -> h in place
  float* y     = BuH  + (size_t)M * 2 * N;      // M*N       (64 MiB)
  float* Bcat  = y    + (size_t)M * N;          // N*2N      (8 MiB)
  float* Ccat  = Bcat + (size_t)N * 2 * N;      // 2N*N      (8 MiB)
  float* lam_r = Ccat + (size_t)2 * N * N;      // N
  float* lam_i = lam_r + N;                     // N
  float* Echk  = lam_i + N;                     // NCHUNK*B*2N (2 MiB)
  float* Schk  = Echk + (size_t)NCHUNK * Bb * 2 * N;  // NCHUNK*B*2N (2 MiB)

  // --- parameter prep ---
  prep_lam<<<(N + 255) / 256, 256, 0, stream>>>(nu_log, theta_log, lam_r, lam_i, N);
  prep_Bcat<<<(N * 2 * N + 255) / 256, 256, 0, stream>>>(B_real, B_imag, gamma_log, Bcat, N);
  prep_Ccat<<<(2 * N * N + 255) / 256, 256, 0, stream>>>(C_real, C_imag, Ccat, N);

  const dim3 blk(256);
  // G1: u = x @ W_in + b_in                 [M,E]@[E,N]
  gemm_f32_wmma<<<dim3(N / TILE_N, M / TILE_M), blk, 0, stream>>>(
      x, W_in, u, b_in, nullptr, nullptr, M, E, N);
  // G2: [Bu_r|Bu_i] = u @ Bcat              [M,N]@[N,2N]
  gemm_f32_wmma<<<dim3(2 * N / TILE_N, M / TILE_M), blk, 0, stream>>>(
      u, Bcat, BuH, nullptr, nullptr, nullptr, M, N, 2 * N);
  // parallel chunked scan (in place: Bu -> h)
  {
    const int p13 = NCHUNK * Bb * N;   // 262144 threads
    scan_partial<<<(p13 + 255) / 256, 256, 0, stream>>>(BuH, lam_r, lam_i, Echk, T, Bb, N);
    scan_starts<<<(Bb * N + 255) / 256, 256, 0, stream>>>(Echk, lam_r, lam_i, Schk, T, Bb, N);
    scan_final<<<(p13 + 255) / 256, 256, 0, stream>>>(BuH, lam_r, lam_i, Schk, T, Bb, N);
  }
  // G3: y = [h_r|h_i] @ [C_r;-C_i] + D*u    [M,2N]@[2N,N]
  gemm_f32_wmma<<<dim3(N / TILE_N, M / TILE_M), blk, 0, stream>>>(
      BuH, Ccat, y, nullptr, Dvec, u, M, 2 * N, N);
  // G4: out = y @ W_out + b_out             [M,N]@[N,E]
  gemm_f32_wmma<<<dim3(E / TILE_N, M / TILE_M), blk, 0, stream>>>(
      y, W_out, (float*)d_out, b_out, nullptr, nullptr, M, N, E);
}